// FullQKAttention_12713103196750
// MI455X (gfx1250) — compile-verified
//
#include <hip/hip_runtime.h>
#include <hip/hip_bf16.h>
#include <math.h>

typedef float v2f __attribute__((ext_vector_type(2)));
typedef float v8f __attribute__((ext_vector_type(8)));

constexpr int D = 64;                 // head dim (fixed by reference)
constexpr float SCALE = 0.125f;       // 64^-0.5
constexpr int LP = 18;                // LDS row pitch (floats): 8B-aligned, conflict-free

// ---------------------------------------------------------------------------
// Kernel A: k_hat = qk / max(||qk||_2, 1e-12), one wave32 per row (d=64 -> 2/lane)
// ---------------------------------------------------------------------------
__global__ void __launch_bounds__(128) qk_norm_kernel(const float* __restrict__ qk,
                                                      float* __restrict__ kn,
                                                      int total_rows) {
  const int lane = threadIdx.x & 31;
  const int wave = threadIdx.x >> 5;
  const int row = blockIdx.x * 4 + wave;
  if (row >= total_rows) return;
  const float2* src = reinterpret_cast<const float2*>(qk + (size_t)row * D);
  float2 x = src[lane];
  float ss = x.x * x.x + x.y * x.y;
  ss += __shfl_xor(ss, 16, 32);
  ss += __shfl_xor(ss, 8, 32);
  ss += __shfl_xor(ss, 4, 32);
  ss += __shfl_xor(ss, 2, 32);
  ss += __shfl_xor(ss, 1, 32);
  const float inv = 1.0f / fmaxf(sqrtf(ss), 1e-12f);
  float2 o;
  o.x = x.x * inv;
  o.y = x.y * inv;
  reinterpret_cast<float2*>(kn + (size_t)row * D)[lane] = o;
}

// ---------------------------------------------------------------------------
// Per-wave body: one 16-row tile starting at r0, streaming causal 16-col tiles.
//   logits = Q K^T * scale   (16x v_wmma_f32_16x16x4_f32, K=64)
//   e = exp(logits)          (diag/-50000 and strict-upper/-inf -> exactly 0)
//   attn <- e (unnormalized); rowsum += e; out_acc += e @ V (WMMA via LDS transpose)
// ---------------------------------------------------------------------------
__device__ __forceinline__ void attn_rowtile(const float* __restrict__ qk,
                                             const float* __restrict__ kn,
                                             const float* __restrict__ vin,
                                             float* __restrict__ out,
                                             float* __restrict__ attn,
                                             float* __restrict__ rsum,
                                             int S, int bi, int r0, int lane,
                                             float* mylds) {
  const int hi = lane >> 4;        // half-wave select
  const int n = lane & 15;         // N index (also M index for A fragments)
  const int kh = hi * 2;           // K sub-offset inside a 16x16x4 step

  // Preload A fragments: Q rows r0..r0+15 (UNnormalized qk), K = 0..63 (b64 loads)
  v2f aq[16];
  {
    const v2f* p2 = reinterpret_cast<const v2f*>(qk + ((size_t)bi * S + r0 + n) * D + kh);
#pragma unroll
    for (int kk = 0; kk < 16; ++kk) aq[kk] = p2[2 * kk];
  }

  v8f oacc[4];
#pragma unroll
  for (int t = 0; t < 4; ++t)
#pragma unroll
    for (int vv = 0; vv < 8; ++vv) oacc[t][vv] = 0.0f;

  float sums[8] = {0.f, 0.f, 0.f, 0.f, 0.f, 0.f, 0.f, 0.f};

  const int nct = r0 / 16 + 1;     // causal column tiles for this row tile
  for (int ct = 0; ct < nct; ++ct) {
    const int c0 = ct * 16;

    // B fragments: k_hat^T tile (cols c0..c0+15, K = 0..63), b64 loads
    v2f bq[16];
    {
      const v2f* kb2 =
          reinterpret_cast<const v2f*>(kn + ((size_t)bi * S + c0 + n) * D + kh);
#pragma unroll
      for (int kk = 0; kk < 16; ++kk) bq[kk] = kb2[2 * kk];
    }
    if (ct + 1 < nct)  // prefetch next K tile -> global_prefetch_b8
      __builtin_prefetch(kn + ((size_t)bi * S + c0 + 16 + n) * D + kh, 0, 1);

    v8f c = {0.f, 0.f, 0.f, 0.f, 0.f, 0.f, 0.f, 0.f};
#pragma unroll
    for (int kk = 0; kk < 16; ++kk)
      c = __builtin_amdgcn_wmma_f32_16x16x4_f32(false, aq[kk], false, bq[kk],
                                                (short)0, c, false, false);

    const bool diag = (ct == nct - 1);
    float e[8];
#pragma unroll
    for (int vv = 0; vv < 8; ++vv) {
      const int m = vv + hi * 8;            // row within tile
      float ev = __expf(c[vv] * SCALE);     // softmax shift-invariant; logits O(1)
      if (diag && n >= m) ev = 0.0f;        // diag(-50000)/strict-upper(-inf) -> 0
      e[vv] = ev;
      sums[vv] += ev;
    }

    // write unnormalized e into attn output region (16 lanes/row -> 64B segments)
    {
      float* arow = attn + ((size_t)bi * S + r0) * S + c0 + n;
#pragma unroll
      for (int vv = 0; vv < 8; ++vv) {
        const int m = vv + hi * 8;
        arow[(size_t)m * S] = e[vv];
      }
    }

    // stage e (C/D layout) into LDS, read back in A layout (same-wave, DScnt-ordered)
#pragma unroll
    for (int vv = 0; vv < 8; ++vv) {
      const int m = vv + hi * 8;
      mylds[m * LP + n] = e[vv];
    }
    {
      const v2f* lp2 = reinterpret_cast<const v2f*>(&mylds[n * LP + kh]);
#pragma unroll
      for (int kk = 0; kk < 4; ++kk) {
        v2f ap = lp2[2 * kk];                 // single ds_load_b64
        const float* vb = vin + ((size_t)bi * S + c0 + 4 * kk + kh) * D;
#pragma unroll
        for (int t = 0; t < 4; ++t) {
          v2f bp;
          bp.x = vb[t * 16 + n];
          bp.y = vb[D + t * 16 + n];
          oacc[t] = __builtin_amdgcn_wmma_f32_16x16x4_f32(false, ap, false, bp,
                                                          (short)0, oacc[t], false, false);
        }
      }
    }
  }

  // reduce row sums across the 16-lane group holding each row
#pragma unroll
  for (int vv = 0; vv < 8; ++vv) {
    float s = sums[vv];
    s += __shfl_xor(s, 1, 32);
    s += __shfl_xor(s, 2, 32);
    s += __shfl_xor(s, 4, 32);
    s += __shfl_xor(s, 8, 32);
    sums[vv] = s;
  }
  if (n == 0) {
#pragma unroll
    for (int vv = 0; vv < 8; ++vv)
      rsum[(size_t)bi * S + r0 + vv + hi * 8] = sums[vv];
  }

  // out = out_acc / rowsum  (degenerate rowsum==0 -> out = v[row], only row 0)
#pragma unroll
  for (int vv = 0; vv < 8; ++vv) {
    const int m = vv + hi * 8;
    const float fac = (sums[vv] > 0.0f) ? 1.0f / sums[vv] : 0.0f;
    const size_t orow = ((size_t)bi * S + r0 + m) * D;
#pragma unroll
    for (int t = 0; t < 4; ++t) {
      float val = oacc[t][vv] * fac;
      if (sums[vv] == 0.0f) val = vin[orow + t * 16 + n];
      out[orow + t * 16 + n] = val;
    }
  }

  // zero-fill strictly acausal columns (>= r0+16); inside the diag tile they were
  // already written as exact 0 by the mask above.
  const int z0 = r0 + 16;
  if (z0 < S) {
    const int nz = S - z0;   // multiple of 16
    for (int m = 0; m < 16; ++m) {
      float4* dst = reinterpret_cast<float4*>(attn + ((size_t)bi * S + r0 + m) * S + z0);
      for (int idx = lane; idx * 4 < nz; idx += 32)
        dst[idx] = make_float4(0.f, 0.f, 0.f, 0.f);
    }
  }
}

// ---------------------------------------------------------------------------
// Kernel B: each wave handles the tile PAIR (p, T-1-p) so every wave streams
// exactly T+1 column tiles -> balanced triangular workload across WGPs.
// ---------------------------------------------------------------------------
__global__ void __launch_bounds__(128) attn_main_kernel(const float* __restrict__ qk,
                                                        const float* __restrict__ kn,
                                                        const float* __restrict__ vin,
                                                        float* __restrict__ out,
                                                        float* __restrict__ attn,
                                                        float* __restrict__ rsum,
                                                        int S) {
  __shared__ float plds[4][16 * LP];   // per-wave padded 16x16 transpose buffer

  const int lane = threadIdx.x & 31;
  const int wave = threadIdx.x >> 5;
  const int blocks_per_b = S / 128;           // 4 waves * 2 tiles * 16 rows per block
  const int bi = blockIdx.x / blocks_per_b;
  const int pair = (blockIdx.x % blocks_per_b) * 4 + wave;   // 0 .. S/32-1
  const int T = S / 16;                       // 16-row tiles per batch image

  float* mylds = &plds[wave][0];
  attn_rowtile(qk, kn, vin, out, attn, rsum, S, bi, 16 * pair, lane, mylds);
  attn_rowtile(qk, kn, vin, out, attn, rsum, S, bi, 16 * (T - 1 - pair), lane, mylds);
}

// ---------------------------------------------------------------------------
// Kernel C: normalize attn in place over the causal span of each row.
// rowsum==0 (row 0): softmax of [-50000, -inf, ...] == one-hot at the diagonal.
// ---------------------------------------------------------------------------
__global__ void __launch_bounds__(256) attn_scale_kernel(float* __restrict__ attn,
                                                         const float* __restrict__ rsum,
                                                         int S) {
  const int bi = blockIdx.x / S;
  const int i = blockIdx.x % S;
  const float s = rsum[(size_t)bi * S + i];
  float* row = attn + ((size_t)bi * S + i) * S;
  if (s > 0.0f) {
    const float rinv = 1.0f / s;
    for (int j = threadIdx.x; j <= i; j += blockDim.x) row[j] *= rinv;
  } else {
    for (int j = threadIdx.x; j <= i; j += blockDim.x) row[j] = (j == i) ? 1.0f : 0.0f;
  }
}

// ---------------------------------------------------------------------------
extern "C" void kernel_launch(void* const* d_in, const int* in_sizes, int n_in,
                              void* d_out, int out_size, void* d_ws, size_t ws_size,
                              hipStream_t stream) {
  const float* qk = (const float*)d_in[0];
  const float* vin = (const float*)d_in[1];
  const int Bn = 4;
  const int S = in_sizes[0] / (Bn * D);   // 4096

  float* out = (float*)d_out;                              // (b, s, d)
  float* attn = (float*)d_out + (size_t)Bn * S * D;        // (b, s, s)
  float* kn = (float*)d_ws;                                // normalized keys
  float* rsum = kn + (size_t)Bn * S * D;                   // per-row exp sums

  const int total_rows = Bn * S;
  qk_norm_kernel<<<(total_rows + 3) / 4, 128, 0, stream>>>(qk, kn, total_rows);
  attn_main_kernel<<<Bn * (S / 128), 128, 0, stream>>>(qk, kn, vin, out, attn, rsum, S);
  attn_scale_kernel<<<Bn * S, 256, 0, stream>>>(attn, rsum, S);
}